// SE3ELayer_61418032332911
// MI455X (gfx1250) — compile-verified
//
#include <hip/hip_runtime.h>
#include <hip/hip_bf16.h>
#include <math.h>

typedef __attribute__((ext_vector_type(2))) float v2f;
typedef __attribute__((ext_vector_type(8))) float v8f;

#define NN 512
#define BB 4
#define HH 16

__device__ inline v8f v8zero() { v8f z = {0.f,0.f,0.f,0.f,0.f,0.f,0.f,0.f}; return z; }

__device__ inline v8f wmma4(v2f a, v2f b, v8f c) {
  // V_WMMA_F32_16X16X4_F32, D = A(16x4) x B(4x16) + C(16x16), fp32 exact
  return __builtin_amdgcn_wmma_f32_16x16x4_f32(false, a, false, b, (short)0, c, false, false);
}

// ---------------- kernel 1: ad = transpose(attn_disp,(0,2,3,1)), inf->0 -----
__global__ __launch_bounds__(256)
void k_transpose(const float* __restrict__ in, float* __restrict__ outp) {
  int id = blockIdx.x;            // b*512*8 + i*8 + tj
  int tj = id & 7;
  int i  = (id >> 3) & 511;
  int b  = id >> 12;
  int t  = threadIdx.x;
  int jj = t & 63, hh = t >> 6;
  __shared__ float tile[16][65];
  int j0 = tj * 64;
#pragma unroll
  for (int s = 0; s < 4; ++s) {
    int h = hh * 4 + s;
    float v = in[(((size_t)b*HH + h)*NN + i)*NN + j0 + jj];
    unsigned u = __float_as_uint(v);
    if ((u & 0x7fffffffu) == 0x7f800000u) v = 0.f;
    tile[h][jj] = v;
  }
  __syncthreads();
  size_t base = (((size_t)b*NN + i)*NN + j0) * 16;
#pragma unroll
  for (int s = 0; s < 4; ++s) {
    int f = t + 256 * s;                // f = jLocal*16 + h  (fully coalesced)
    outp[base + f] = tile[f & 15][f >> 4];
  }
}

// ---------------- kernel 2: normalizer rows ---------------------------------
__global__ __launch_bounds__(128)
void k_norm(const float* __restrict__ attn, float* __restrict__ normv) {
  int id = blockIdx.x;                 // b*512 + i
  int b = id >> 9, i = id & 511;
  int t = threadIdx.x;
  const float* row = attn + ((size_t)b*HH*NN + i)*NN;   // attn_disp[b,0,i,:]
  float cnt = 0.f;
#pragma unroll
  for (int c = 0; c < 4; ++c) {
    float v = row[t*4 + c];
    unsigned u = __float_as_uint(v);
    cnt += ((u & 0x7fffffffu) == 0x7f800000u) ? 0.f : 1.f;
  }
  __shared__ float red[128];
  red[t] = cnt;
  __syncthreads();
  for (int s = 64; s > 0; s >>= 1) {
    if (t < s) red[t] += red[t + s];
    __syncthreads();
  }
  if (t == 0) normv[id] = sqrtf(red[0]);
}

// ---------------- kernel 3: edge features (l=0,1 spherical) -----------------
__global__ __launch_bounds__(256)
void k_ef(const float* __restrict__ coords, float* __restrict__ efp) {
  int e = blockIdx.x * 256 + threadIdx.x;   // b,i,j flattened
  int b = e >> 18;
  int i = (e >> 9) & 511;
  int j = e & 511;
  const float* ci = coords + ((size_t)b*NN + i)*3;
  const float* cj = coords + ((size_t)b*NN + j)*3;
  float dx = ci[0]-cj[0], dy = ci[1]-cj[1], dz = ci[2]-cj[2];
  float r = sqrtf(dx*dx + dy*dy + dz*dz + 1e-12f);
  float inv1 = 1.0f / (r + 1e-5f);
  dx *= inv1; dy *= inv1; dz *= inv1;
  float n = sqrtf(dx*dx + dy*dy + dz*dz + 1e-12f);
  float s3n = 1.7320508075688772f / n;
  float4 o;
  o.x = 1.0f;
  o.y = s3n * dy;   // u[1]
  o.z = s3n * dz;   // u[2]
  o.w = s3n * dx;   // u[0]
  reinterpret_cast<float4*>(efp)[e] = o;
}

// ---------------- kernel 4: nf0[b,j] = sum_i ef[b,i,j]/norm[b,j] ------------
__global__ __launch_bounds__(256)
void k_nf0(const float* __restrict__ efp, const float* __restrict__ normv,
           float* __restrict__ nf0) {
  int id = blockIdx.x;                 // b*8 + jt
  int b = id >> 3, jt = id & 7;
  int t = threadIdx.x;
  int jl = t & 63, strip = t >> 6;
  int j = jt*64 + jl;
  float ax=0.f, ay=0.f, az=0.f, aw=0.f;
  for (int i = strip; i < NN; i += 4) {
    float4 v = reinterpret_cast<const float4*>(efp)[((size_t)b*NN + i)*NN + j];
    ax += v.x; ay += v.y; az += v.z; aw += v.w;
  }
  __shared__ float red[4][256];
  red[0][t]=ax; red[1][t]=ay; red[2][t]=az; red[3][t]=aw;
  __syncthreads();
  if (t < 64) {
    float invn = 1.0f / normv[b*NN + j];
    float4 o;
    o.x = (red[0][t]+red[0][t+64]+red[0][t+128]+red[0][t+192]) * invn;
    o.y = (red[1][t]+red[1][t+64]+red[1][t+128]+red[1][t+192]) * invn;
    o.z = (red[2][t]+red[2][t+64]+red[2][t+128]+red[2][t+192]) * invn;
    o.w = (red[3][t]+red[3][t+64]+red[3][t+128]+red[3][t+192]) * invn;
    reinterpret_cast<float4*>(nf0)[b*NN + j] = o;
  }
}

// ---------------- kernel 5: fused conv1 -> gate -> conv2 (WMMA core) --------
// One wave per (b,j). For chunks of 16 i-rows: W = FC(ad rows) via
// v_wmma_f32_16x16x4_f32, accumulate moments T[k][n] = sum_i u_i[k]*w_i[n],
// then a scalar equivariant combine.
__global__ __launch_bounds__(32)
void k_conv(const float* __restrict__ ad,      // (B,N,N,16)
            const float* __restrict__ efp,     // (B,N,N,4)
            const float* __restrict__ nf0,     // (B,N,4)
            const float* __restrict__ normv,   // (B,N)
            const float* __restrict__ w1a, const float* __restrict__ w2a,
            const float* __restrict__ w1b, const float* __restrict__ w2b,
            float* __restrict__ nf_out)        // (B,N,6)
{
  const int bj = blockIdx.x;
  const int b  = bj >> 9;
  const int j  = bj & 511;
  const int L  = threadIdx.x;
  const int hi = L >> 4;
  const int lo = L & 15;

  __shared__ float ldsA[16][20];
  __shared__ float ldsH[16][36];
  __shared__ float ldsU[16][4];
  __shared__ float ldsT1[4][16];
  __shared__ float ldsT2[4][16];

  // Preload weight B-fragments (scale 1/sqrt(16) and sqrt(2)/sqrt(32) fold to 0.25 each)
  v2f b1a[2][4], b1b[2][4], b2a[8], b2b[8];
#pragma unroll
  for (int nt = 0; nt < 2; ++nt)
#pragma unroll
    for (int s = 0; s < 4; ++s) {
      int k0 = 4*s + 2*hi;
      int n = nt*16 + lo;
      b1a[nt][s].x = 0.25f * w1a[k0*32 + n];
      b1a[nt][s].y = 0.25f * w1a[(k0+1)*32 + n];
      b1b[nt][s].x = 0.25f * w1b[k0*32 + n];
      b1b[nt][s].y = 0.25f * w1b[(k0+1)*32 + n];
    }
#pragma unroll
  for (int s = 0; s < 8; ++s) {
    int k0 = 4*s + 2*hi;
    b2a[s].x = (lo < 14) ? 0.25f * w2a[k0*14 + lo]     : 0.f;
    b2a[s].y = (lo < 14) ? 0.25f * w2a[(k0+1)*14 + lo] : 0.f;
    b2b[s].x = (lo < 12) ? 0.25f * w2b[k0*12 + lo]     : 0.f;
    b2b[s].y = (lo < 12) ? 0.25f * w2b[(k0+1)*12 + lo] : 0.f;
  }

  float acc1[4] = {0.f,0.f,0.f,0.f};
  float acc2[4] = {0.f,0.f,0.f,0.f};

#pragma unroll 1
  for (int i0 = 0; i0 < NN; i0 += 16) {
    // ---- stage A tile (16 edges x 16 heads) and U = ef rows ----
#pragma unroll
    for (int tt = 0; tt < 2; ++tt) {
      int idx = L + 32*tt;                 // 0..63
      int row = idx >> 2, c4 = (idx & 3) * 4;
      const float* src = ad + ((((size_t)b*NN + (i0+row))*NN + j) << 4) + c4;
      float4 v = *reinterpret_cast<const float4*>(src);
      ldsA[row][c4+0] = v.x; ldsA[row][c4+1] = v.y;
      ldsA[row][c4+2] = v.z; ldsA[row][c4+3] = v.w;
    }
    if (L < 16) {
      float4 v = reinterpret_cast<const float4*>(efp)[((size_t)b*NN + (i0+L))*NN + j];
      ldsU[L][0] = v.x; ldsU[L][1] = v.y; ldsU[L][2] = v.z; ldsU[L][3] = v.w;
    }
    __syncthreads();

    // ---- net1 layer1: h1 = relu(A @ 0.25*W1a), two 16-col tiles ----
    v8f c1x = v8zero(), c1y = v8zero();
#pragma unroll
    for (int s = 0; s < 4; ++s) {
      v2f a; a.x = ldsA[lo][4*s + 2*hi]; a.y = ldsA[lo][4*s + 2*hi + 1];
      c1x = wmma4(a, b1a[0][s], c1x);
      c1y = wmma4(a, b1a[1][s], c1y);
    }
#pragma unroll
    for (int e = 0; e < 8; ++e) {
      float x = c1x[e]; ldsH[e + 8*hi][lo]      = x > 0.f ? x : 0.f;
      float y = c1y[e]; ldsH[e + 8*hi][16 + lo] = y > 0.f ? y : 0.f;
    }
    __syncthreads();

    // ---- net1 layer2: w = h1 @ 0.25*W2a ----
    v8f c2 = v8zero();
#pragma unroll
    for (int s = 0; s < 8; ++s) {
      v2f a; a.x = ldsH[lo][4*s + 2*hi]; a.y = ldsH[lo][4*s + 2*hi + 1];
      c2 = wmma4(a, b2a[s], c2);
    }
#pragma unroll
    for (int e = 0; e < 8; ++e) {
      int m = e + 8*hi;
      float w = c2[e];
      acc1[0] += ldsU[m][0] * w;
      acc1[1] += ldsU[m][1] * w;
      acc1[2] += ldsU[m][2] * w;
      acc1[3] += ldsU[m][3] * w;
    }

    // ---- net2 layer1 (reuse A tile) ----
    v8f d1x = v8zero(), d1y = v8zero();
#pragma unroll
    for (int s = 0; s < 4; ++s) {
      v2f a; a.x = ldsA[lo][4*s + 2*hi]; a.y = ldsA[lo][4*s + 2*hi + 1];
      d1x = wmma4(a, b1b[0][s], d1x);
      d1y = wmma4(a, b1b[1][s], d1y);
    }
    __syncthreads();     // everyone done reading ldsH (net1 L2)
#pragma unroll
    for (int e = 0; e < 8; ++e) {
      float x = d1x[e]; ldsH[e + 8*hi][lo]      = x > 0.f ? x : 0.f;
      float y = d1y[e]; ldsH[e + 8*hi][16 + lo] = y > 0.f ? y : 0.f;
    }
    __syncthreads();

    // ---- net2 layer2 ----
    v8f d2 = v8zero();
#pragma unroll
    for (int s = 0; s < 8; ++s) {
      v2f a; a.x = ldsH[lo][4*s + 2*hi]; a.y = ldsH[lo][4*s + 2*hi + 1];
      d2 = wmma4(a, b2b[s], d2);
    }
#pragma unroll
    for (int e = 0; e < 8; ++e) {
      int m = e + 8*hi;
      float w = d2[e];
      acc2[0] += ldsU[m][0] * w;
      acc2[1] += ldsU[m][1] * w;
      acc2[2] += ldsU[m][2] * w;
      acc2[3] += ldsU[m][3] * w;
    }
    __syncthreads();     // protect ldsA/ldsU/ldsH for next chunk
  }

  // combine wave halves (rows 0..7 + rows 8..15 of each column)
#pragma unroll
  for (int k = 0; k < 4; ++k) {
    acc1[k] += __shfl_xor(acc1[k], 16);
    acc2[k] += __shfl_xor(acc2[k], 16);
  }
  if (L < 16) {
#pragma unroll
    for (int k = 0; k < 4; ++k) { ldsT1[k][L] = acc1[k]; ldsT2[k][L] = acc2[k]; }
  }
  __syncthreads();

  if (L == 0) {
    const float IS2  = 0.70710678118654752f;  // 1/sqrt(2)
    const float IS6  = 0.40824829046386302f;  // 1/sqrt(6)
    const float IS12 = 0.28867513459481288f;  // 1/sqrt(12)
    float invn = 1.0f / normv[bj];
    const float* x1p = nf0 + 4*bj;
    float s1 = x1p[0], v1x = x1p[1], v1y = x1p[2], v1z = x1p[3];

    // conv1 output (20 comps of IR_INTER)
    float o[20];
#pragma unroll
    for (int c = 0; c < 4; ++c)
      o[c] = invn * (IS2*s1*ldsT1[0][c]
                   + IS6*(v1x*ldsT1[1][4+c] + v1y*ldsT1[2][4+c] + v1z*ldsT1[3][4+c]));
    o[4]=0.f; o[5]=0.f; o[6]=0.f; o[7]=0.f;
#pragma unroll
    for (int wi = 0; wi < 2; ++wi) {
      float t0 = ldsT1[0][10+wi];
      o[8+3*wi+0] = invn*IS2*(s1*ldsT1[1][8+wi] + v1x*t0);
      o[8+3*wi+1] = invn*IS2*(s1*ldsT1[2][8+wi] + v1y*t0);
      o[8+3*wi+2] = invn*IS2*(s1*ldsT1[3][8+wi] + v1z*t0);
      float tx = ldsT1[1][12+wi], ty = ldsT1[2][12+wi], tz = ldsT1[3][12+wi];
      o[14+3*wi+0] = invn*IS2*(v1y*tz - v1z*ty);
      o[14+3*wi+1] = invn*IS2*(v1z*tx - v1x*tz);
      o[14+3*wi+2] = invn*IS2*(v1x*ty - v1y*tx);
    }

    // gate
    float se0 = fmaxf(o[0],0.f), se1 = fmaxf(o[1],0.f);
    float g1  = fmaxf(o[2],0.f), g3  = fmaxf(o[3],0.f);
    float so0 = fabsf(o[4]),     so1 = fabsf(o[5]);
    float g2  = tanhf(o[6]),     g4  = tanhf(o[7]);
    float vo0x=o[8]*g1,  vo0y=o[9]*g1,  vo0z=o[10]*g1;
    float vo1x=o[11]*g2, vo1y=o[12]*g2, vo1z=o[13]*g2;
    float ve0x=o[14]*g3, ve0y=o[15]*g3, ve0z=o[16]*g3;
    float ve1x=o[17]*g4, ve1y=o[18]*g4, ve1z=o[19]*g4;

    // conv2 combine (6 comps of IR_OUT)
    float t4x=ldsT2[1][4], t4y=ldsT2[2][4], t4z=ldsT2[3][4];
    float t5x=ldsT2[1][5], t5y=ldsT2[2][5], t5z=ldsT2[3][5];
    float t8x=ldsT2[1][8], t8y=ldsT2[2][8], t8z=ldsT2[3][8];
    float t9x=ldsT2[1][9], t9y=ldsT2[2][9], t9z=ldsT2[3][9];
    float T02=ldsT2[0][2], T03=ldsT2[0][3], T010=ldsT2[0][10], T011=ldsT2[0][11];

    float r0 = invn*( IS6*(se0*ldsT2[1][0] + se1*ldsT2[1][1]) + IS6*(vo0x*T02 + vo1x*T03)
                    + IS12*((ve0y*t4z - ve0z*t4y) + (ve1y*t5z - ve1z*t5y)) );
    float r1 = invn*( IS6*(se0*ldsT2[2][0] + se1*ldsT2[2][1]) + IS6*(vo0y*T02 + vo1y*T03)
                    + IS12*((ve0z*t4x - ve0x*t4z) + (ve1z*t5x - ve1x*t5z)) );
    float r2 = invn*( IS6*(se0*ldsT2[3][0] + se1*ldsT2[3][1]) + IS6*(vo0z*T02 + vo1z*T03)
                    + IS12*((ve0x*t4y - ve0y*t4x) + (ve1x*t5y - ve1y*t5x)) );
    float r3 = invn*( IS6*(so0*ldsT2[1][6] + so1*ldsT2[1][7])
                    + IS12*((vo0y*t8z - vo0z*t8y) + (vo1y*t9z - vo1z*t9y))
                    + IS6*(ve0x*T010 + ve1x*T011) );
    float r4 = invn*( IS6*(so0*ldsT2[2][6] + so1*ldsT2[2][7])
                    + IS12*((vo0z*t8x - vo0x*t8z) + (vo1z*t9x - vo1x*t9z))
                    + IS6*(ve0y*T010 + ve1y*T011) );
    float r5 = invn*( IS6*(so0*ldsT2[3][6] + so1*ldsT2[3][7])
                    + IS12*((vo0x*t8y - vo0y*t8x) + (vo1x*t9y - vo1y*t9x))
                    + IS6*(ve0z*T010 + ve1z*T011) );

    float* op = nf_out + 6*bj;
    op[0]=r0; op[1]=r1; op[2]=r2; op[3]=r3; op[4]=r4; op[5]=r5;
  }
}

// ---------------- host-side launch ------------------------------------------
extern "C" void kernel_launch(void* const* d_in, const int* in_sizes, int n_in,
                              void* d_out, int out_size, void* d_ws, size_t ws_size,
                              hipStream_t stream) {
  const float* attn_disp = (const float*)d_in[0];   // (4,16,512,512)
  const float* attn_mask = (const float*)d_in[1];   // (64,512,512)
  const float* coords    = (const float*)d_in[2];   // (4,512,3)
  const float* fc1_w1    = (const float*)d_in[3];   // (16,32)
  const float* fc1_w2    = (const float*)d_in[4];   // (32,14)
  const float* fc2_w1    = (const float*)d_in[5];   // (16,32)
  const float* fc2_w2    = (const float*)d_in[6];   // (32,12)

  const size_t AD_E    = (size_t)BB*HH*NN*NN;       // 16777216
  const size_t MASK_E  = (size_t)BB*HH*NN*NN;       // 16777216
  const size_t COORD_E = (size_t)BB*NN*3;           // 6144
  const size_t NF_E    = (size_t)BB*NN*6;           // 12288

  float* out     = (float*)d_out;
  float* o_ad    = out;
  float* o_mask  = out + AD_E;
  float* o_coord = out + AD_E + MASK_E;
  float* o_nf    = out + AD_E + MASK_E + COORD_E;
  float* o_ef    = out + AD_E + MASK_E + COORD_E + NF_E;

  float* ws      = (float*)d_ws;
  float* ws_norm = ws;                  // B*N      = 2048
  float* ws_nf0  = ws + BB*NN;          // B*N*4    = 8192

  // pass-through outputs
  hipMemcpyAsync(o_mask, attn_mask, MASK_E * sizeof(float),
                 hipMemcpyDeviceToDevice, stream);
  hipMemcpyAsync(o_coord, coords, COORD_E * sizeof(float),
                 hipMemcpyDeviceToDevice, stream);

  k_transpose<<<BB*NN*(NN/64), 256, 0, stream>>>(attn_disp, o_ad);
  k_norm<<<BB*NN, 128, 0, stream>>>(attn_disp, ws_norm);
  k_ef<<<(BB*NN*NN)/256, 256, 0, stream>>>(coords, o_ef);
  k_nf0<<<BB*(NN/64), 256, 0, stream>>>(o_ef, ws_norm, ws_nf0);
  k_conv<<<BB*NN, 32, 0, stream>>>(o_ad, o_ef, ws_nf0, ws_norm,
                                   fc1_w1, fc1_w2, fc2_w1, fc2_w2, o_nf);
}